// MGHIF_67130338836548
// MI455X (gfx1250) — compile-verified
//
#include <hip/hip_runtime.h>

// MI455X / gfx1250: memory-bound fused path-attention kernel.
// cross@W GEMM runs on V_WMMA_F32_16X16X4_F32 (fp32 data -> fp32 WMMA path).
// W is staged in LDS pair-interleaved so every B fragment is one ds_load_b64.

#define E 64
#define NPATH 5
#define PATHS_PER_BLOCK 16
#define ROWS_PER_BLOCK (PATHS_PER_BLOCK * NPATH)   // 80
#define THREADS 256                                 // 8 wave32

typedef float v2f __attribute__((ext_vector_type(2)));
typedef float v8f __attribute__((ext_vector_type(8)));

__global__ __launch_bounds__(THREADS)
void MGHIF_pathfuse_kernel(const float* __restrict__ U,   // [M,5,64] upaths_emb
                           const float* __restrict__ I,   // [M,5,64] ipaths_emb
                           const float* __restrict__ ST,  // [S,64]   score_table
                           const float* __restrict__ W,   // [64,64]
                           const int*   __restrict__ UP,  // [M,5]
                           const int*   __restrict__ IP,  // [M,5]
                           float* __restrict__ OUT)       // [M,5,64]
{
    // B-fragment friendly W: sWb[kp][c] = { W[2kp][c], W[2kp+1][c] }  (16 KB)
    __shared__ __align__(16) float sWb[E / 2][E][2];
    __shared__ __align__(16) float sX[ROWS_PER_BLOCK][E];   // cross -> wc (20 KB)
    __shared__ __align__(16) float sU[PATHS_PER_BLOCK][3][E]; // u rows 0..2 (12 KB)
    __shared__ __align__(16) float sI[PATHS_PER_BLOCK][3][E]; // i rows 0..2 (12 KB)
    __shared__ __align__(16) float sS[PATHS_PER_BLOCK][3][E]; // score A,B,C (12 KB)

    const int tid   = threadIdx.x;
    const int path0 = blockIdx.x * PATHS_PER_BLOCK;
    const long long row0 = (long long)path0 * NPATH;

    // ---------------- Phase 1: cooperative staging ----------------
    // W: load float4 (coalesced), scatter into pair-interleaved LDS layout.
    for (int i = tid; i < (E * E) / 4; i += THREADS) {
        float4 w4 = ((const float4*)W)[i];
        int k = i / (E / 4);          // W row
        int c = (i % (E / 4)) * 4;    // W col base
        int kp = k >> 1, lo = k & 1;
        sWb[kp][c + 0][lo] = w4.x;
        sWb[kp][c + 1][lo] = w4.y;
        sWb[kp][c + 2][lo] = w4.z;
        sWb[kp][c + 3][lo] = w4.w;
    }

    // u/i rows: build cross = u*i, keep rows 0..2 for the window sums.
    for (int i = tid; i < (ROWS_PER_BLOCK * E) / 4; i += THREADS) {
        int row = i / (E / 4);
        int c4  = i % (E / 4);
        long long g = (row0 + row) * E + c4 * 4;
        float4 u4 = *(const float4*)(U + g);
        float4 i4 = *(const float4*)(I + g);
        float4 x;
        x.x = u4.x * i4.x; x.y = u4.y * i4.y; x.z = u4.z * i4.z; x.w = u4.w * i4.w;
        *(float4*)&sX[row][c4 * 4] = x;
        int rp = row % NPATH;
        if (rp < 3) {
            int p = row / NPATH;
            *(float4*)&sU[p][rp][c4 * 4] = u4;
            *(float4*)&sI[p][rp][c4 * 4] = i4;
        }
    }

    // score-table gathers: per path only rows st[up0], st[up2], st[ip1] are needed
    for (int i = tid; i < PATHS_PER_BLOCK * 3 * (E / 4); i += THREADS) {
        int p    = i / (3 * (E / 4));
        int rem  = i % (3 * (E / 4));
        int slot = rem / (E / 4);
        int c4   = rem % (E / 4);
        int m    = path0 + p;
        int idx  = (slot == 0) ? UP[m * NPATH + 0]
                 : (slot == 1) ? UP[m * NPATH + 2]
                               : IP[m * NPATH + 1];
        *(float4*)&sS[p][slot][c4 * 4] =
            *(const float4*)(ST + (long long)idx * E + c4 * 4);
    }
    __syncthreads();

    // ---------------- Phase 2: wc = cross @ W via V_WMMA_F32_16X16X4_F32 ----------------
    const int wave = tid >> 5;
    const int lane = tid & 31;
    if (wave < ROWS_PER_BLOCK / 16) {      // 5 waves own one 16-row tile each
        const int tileRow = wave * 16;
        const int r    = lane & 15;        // matrix row (A) / column (B,D)
        const int half = lane >> 4;        // K-pair select: 0 -> K{0,1}, 1 -> K{2,3}
        v8f acc[4];
        #pragma unroll
        for (int n = 0; n < 4; ++n) {      // 4 column tiles of 16
            v8f c = {};
            #pragma unroll
            for (int kc = 0; kc < 16; ++kc) {   // K = 64 in chunks of 4
                const int k0 = 4 * kc + 2 * half;      // even -> 8B aligned
                v2f a = *(const v2f*)&sX[tileRow + r][k0];          // ds_load_b64
                v2f b = *(const v2f*)&sWb[k0 >> 1][n * 16 + r][0];  // ds_load_b64
                c = __builtin_amdgcn_wmma_f32_16x16x4_f32(
                        false, a, false, b, (short)0, c, false, false);
            }
            acc[n] = c;
        }
        // D layout: VGPR v, lanes 0-15 -> M=v, lanes 16-31 -> M=v+8. Overwrite cross.
        #pragma unroll
        for (int n = 0; n < 4; ++n)
            #pragma unroll
            for (int v = 0; v < 8; ++v)
                sX[tileRow + v + 8 * half][n * 16 + r] = acc[n][v];
    }
    __syncthreads();

    // ---------------- Phase 3: logits, softmax over 3, blended output ----------------
    if (tid < 2 * ROWS_PER_BLOCK) {        // 2 threads per row, 32 elems each
        const int row = tid >> 1;
        const int h   = tid & 1;
        const int p   = row / NPATH;
        const int j   = row % NPATH;
        // window combination weights over (row0,row1,row2) building blocks
        const float WUt[5][3] = {{1,0,0},{1,1,0},{1,1,1},{0,1,1},{0,0,1}}; // pus
        const float WIt[5][3] = {{0,0,1},{0,1,1},{1,1,1},{1,1,0},{1,0,0}}; // pis
        const float WSt[5][3] = {{1,0,0},{1,0,1},{1,1,1},{0,1,1},{0,1,0}}; // pss (A,B,C)
        const float wu0 = WUt[j][0], wu1 = WUt[j][1], wu2 = WUt[j][2];
        const float wi0 = WIt[j][0], wi1 = WIt[j][1], wi2 = WIt[j][2];
        const float ws0 = WSt[j][0], ws1 = WSt[j][1], ws2 = WSt[j][2];

        float l0 = 0.f, l1 = 0.f, l2 = 0.f;
        const int eBase = h * 32;
        for (int e = eBase; e < eBase + 32; ++e) {
            float a0 = wu0 * sU[p][0][e] + wu1 * sU[p][1][e] + wu2 * sU[p][2][e];
            float a1 = wi0 * sI[p][0][e] + wi1 * sI[p][1][e] + wi2 * sI[p][2][e];
            float a2 = ws0 * sS[p][0][e] + ws1 * sS[p][1][e] + ws2 * sS[p][2][e];
            float w  = sX[row][e];                 // wc
            l0 += a0 * w; l1 += a1 * w; l2 += a2 * w;
        }
        // pair (tid, tid^1) lives in the same wave32 -> combine halves
        l0 += __shfl_xor(l0, 1, 32);
        l1 += __shfl_xor(l1, 1, 32);
        l2 += __shfl_xor(l2, 1, 32);

        float mx = fmaxf(l0, fmaxf(l1, l2));
        float x0 = __expf(l0 - mx), x1 = __expf(l1 - mx), x2 = __expf(l2 - mx);
        float inv = 1.0f / (x0 + x1 + x2);
        float t0 = x0 * inv, t1 = x1 * inv, t2 = x2 * inv;

        float* outp = OUT + (row0 + row) * E;
        for (int e = eBase; e < eBase + 32; ++e) {
            float a0 = wu0 * sU[p][0][e] + wu1 * sU[p][1][e] + wu2 * sU[p][2][e];
            float a1 = wi0 * sI[p][0][e] + wi1 * sI[p][1][e] + wi2 * sI[p][2][e];
            float a2 = ws0 * sS[p][0][e] + ws1 * sS[p][1][e] + ws2 * sS[p][2][e];
            outp[e] = t0 * a0 + t1 * a1 + t2 * a2;
        }
    }
}

extern "C" void kernel_launch(void* const* d_in, const int* in_sizes, int n_in,
                              void* d_out, int out_size, void* d_ws, size_t ws_size,
                              hipStream_t stream) {
    (void)n_in; (void)d_ws; (void)ws_size; (void)out_size;
    const float* U  = (const float*)d_in[0];   // upaths_emb [M,5,64]
    const float* I  = (const float*)d_in[1];   // ipaths_emb [M,5,64]
    const float* ST = (const float*)d_in[2];   // score_table [S,64]
    const float* W  = (const float*)d_in[3];   // W [64,64]
    const int*   UP = (const int*)d_in[4];     // upaths [M,5]
    const int*   IP = (const int*)d_in[5];     // ipaths [M,5]
    float* OUT = (float*)d_out;

    const int M = in_sizes[0] / (NPATH * E);   // 131072
    const int blocks = M / PATHS_PER_BLOCK;    // 8192 (M divisible by 16)
    MGHIF_pathfuse_kernel<<<blocks, THREADS, 0, stream>>>(U, I, ST, W, UP, IP, OUT);
}